// Attention_75265006895865
// MI455X (gfx1250) — compile-verified
//
#include <hip/hip_runtime.h>
#include <hip/hip_bf16.h>
#include <math.h>

#define B_   256
#define AREA 196
#define C_   2048
#define ATT  512
#define RNN  512

typedef __attribute__((ext_vector_type(2))) float v2f;
typedef __attribute__((ext_vector_type(8))) float v8f;

// ---------------------------------------------------------------------------
// fp32 WMMA GEMM:  D[m][n] = act( sum_k A[m][k] * Bt[n][k] + bias[n] )
// A: row-major M x K.  Bt: row-major N x K (how W1 (ATT,RNN) and W3 (C,2C)
// are already stored for the reference's  x @ W^T  products).
//
// One wave32 computes a 32x64 output tile as a 2x4 grid of 16x16 WMMA tiles:
// per K-step of 4 it loads 2 A-fragments + 4 B-fragments (6 x b64 loads) and
// issues 8 V_WMMA_F32_16X16X4_F32 — ~2.7x the arithmetic intensity of a single
// 16x16 tile, cutting L2 operand traffic from ~1 GB to ~0.4 GB for GEMM #2.
// Requires M%32==0, N%64==0, K%4==0, grid covers exactly (M/32)*(N/64) waves.
// ---------------------------------------------------------------------------
__global__ __launch_bounds__(256) void wmma_gemm_atb(
    const float* __restrict__ A, const float* __restrict__ Bt,
    const float* __restrict__ bias, float* __restrict__ D,
    int N, int K, int tiles_n /* = N/64 */, int relu)
{
  const int wave = (blockIdx.x * blockDim.x + threadIdx.x) >> 5;
  const int lane = threadIdx.x & 31;
  const int mt = wave / tiles_n;    // 32-row tile index
  const int nt = wave % tiles_n;    // 64-col tile index
  const int half = lane >> 4;       // 0: lanes 0-15, 1: lanes 16-31
  const int r    = lane & 15;

  // A frag per ISA 16x4 f32 layout: lane half selects K pair {0,1} vs {2,3}
  const float* a0 = A  + (size_t)(mt * 32 + r) * K + half * 2;
  const float* a1 = a0 + (size_t)16 * K;
  const float* b0 = Bt + (size_t)(nt * 64 + r) * K + half * 2;
  const float* b1 = b0 + (size_t)16 * K;
  const float* b2 = b0 + (size_t)32 * K;
  const float* b3 = b0 + (size_t)48 * K;

  v8f c[8] = {};
  for (int k = 0; k < K; k += 4) {
    v2f A0 = *(const v2f*)(a0 + k);
    v2f A1 = *(const v2f*)(a1 + k);
    v2f B0 = *(const v2f*)(b0 + k);
    v2f B1 = *(const v2f*)(b1 + k);
    v2f B2 = *(const v2f*)(b2 + k);
    v2f B3 = *(const v2f*)(b3 + k);
    c[0] = __builtin_amdgcn_wmma_f32_16x16x4_f32(false, A0, false, B0, (short)0, c[0], false, false);
    c[1] = __builtin_amdgcn_wmma_f32_16x16x4_f32(false, A0, false, B1, (short)0, c[1], false, false);
    c[2] = __builtin_amdgcn_wmma_f32_16x16x4_f32(false, A0, false, B2, (short)0, c[2], false, false);
    c[3] = __builtin_amdgcn_wmma_f32_16x16x4_f32(false, A0, false, B3, (short)0, c[3], false, false);
    c[4] = __builtin_amdgcn_wmma_f32_16x16x4_f32(false, A1, false, B0, (short)0, c[4], false, false);
    c[5] = __builtin_amdgcn_wmma_f32_16x16x4_f32(false, A1, false, B1, (short)0, c[5], false, false);
    c[6] = __builtin_amdgcn_wmma_f32_16x16x4_f32(false, A1, false, B2, (short)0, c[6], false, false);
    c[7] = __builtin_amdgcn_wmma_f32_16x16x4_f32(false, A1, false, B3, (short)0, c[7], false, false);
  }

  // C/D layout: VGPR v of each accumulator holds row (v + half*8), col = r
  #pragma unroll
  for (int im = 0; im < 2; ++im) {
    #pragma unroll
    for (int jn = 0; jn < 4; ++jn) {
      const v8f cc = c[im * 4 + jn];
      const int col = nt * 64 + jn * 16 + r;
      const float bv = bias[col];
      #pragma unroll
      for (int v = 0; v < 8; ++v) {
        float val = cc[v] + bv;
        if (relu) val = fmaxf(val, 0.0f);
        D[(size_t)(mt * 32 + im * 16 + half * 8 + v) * N + col] = val;
      }
    }
  }
}

// ---------------------------------------------------------------------------
// logits[b,s] = sum_a tanh(att[b,s,a] + z1[b,a]) * W2[a]  + b2
// One wave32 per (b,s) row; float4 loads; xor-shuffle wave reduction.
// ---------------------------------------------------------------------------
__global__ __launch_bounds__(256) void logits_kernel(
    const float* __restrict__ att, const float* __restrict__ z1,
    const float* __restrict__ W2, const float* __restrict__ b2,
    float* __restrict__ logits)
{
  const int wave = (blockIdx.x * blockDim.x + threadIdx.x) >> 5;
  const int lane = threadIdx.x & 31;
  if (wave >= B_ * AREA) return;
  const int b = wave / AREA;

  const float4* ap = (const float4*)(att + (size_t)wave * ATT);
  const float4* zp = (const float4*)(z1 + (size_t)b * ATT);
  const float4* wp = (const float4*)W2;

  float acc = 0.0f;
  #pragma unroll
  for (int i = lane; i < ATT / 4; i += 32) {
    float4 av = ap[i], zv = zp[i], wv = wp[i];
    acc += tanhf(av.x + zv.x) * wv.x;
    acc += tanhf(av.y + zv.y) * wv.y;
    acc += tanhf(av.z + zv.z) * wv.z;
    acc += tanhf(av.w + zv.w) * wv.w;
  }
  #pragma unroll
  for (int off = 16; off > 0; off >>= 1)
    acc += __shfl_xor(acc, off, 32);
  if (lane == 0) logits[wave] = acc + b2[0];
}

// ---------------------------------------------------------------------------
// Row softmax over AREA per batch. Writes attention weights `a` (2nd output).
// ---------------------------------------------------------------------------
__global__ __launch_bounds__(256) void softmax_kernel(
    const float* __restrict__ logits, float* __restrict__ a_out)
{
  __shared__ float redm[8];
  __shared__ float reds[8];
  const int b = blockIdx.x;
  const int t = threadIdx.x;
  const int lane = t & 31, w = t >> 5;

  float v = (t < AREA) ? logits[b * AREA + t] : -INFINITY;

  float m = v;
  #pragma unroll
  for (int off = 16; off > 0; off >>= 1)
    m = fmaxf(m, __shfl_xor(m, off, 32));
  if (lane == 0) redm[w] = m;
  __syncthreads();
  m = redm[0];
  #pragma unroll
  for (int i = 1; i < 8; ++i) m = fmaxf(m, redm[i]);

  float e = (t < AREA) ? __expf(v - m) : 0.0f;
  float s = e;
  #pragma unroll
  for (int off = 16; off > 0; off >>= 1)
    s += __shfl_xor(s, off, 32);
  if (lane == 0) reds[w] = s;
  __syncthreads();
  s = reds[0];
  #pragma unroll
  for (int i = 1; i < 8; ++i) s += reds[i];

  if (t < AREA) a_out[b * AREA + t] = e / s;
}

// ---------------------------------------------------------------------------
// emb[b, c]      = sum_s img[b,s,c] * a[b,s]      (self_features)
// emb[b, C + c]  = sum_s img[b,s,c] * a_cap[b,s]  (cap_features)
// One block per batch row; thread t owns channels [8t, 8t+8); the whole block
// streams each 8 KB spatial row of img_features with contiguous float4 loads.
// ---------------------------------------------------------------------------
__global__ __launch_bounds__(256) void pool_kernel(
    const float* __restrict__ img, const float* __restrict__ a,
    const float* __restrict__ acap, float* __restrict__ emb)
{
  const int b = blockIdx.x;
  const int c0 = threadIdx.x * 8;
  const float* base = img + (size_t)b * AREA * C_ + c0;

  float sf[8] = {0,0,0,0,0,0,0,0};
  float cf[8] = {0,0,0,0,0,0,0,0};

  for (int s = 0; s < AREA; ++s) {
    const float wa = a[b * AREA + s];
    const float wc = acap[b * AREA + s];
    const float4* p = (const float4*)(base + (size_t)s * C_);
    float4 x0 = p[0];
    float4 x1 = p[1];
    sf[0] += x0.x * wa; cf[0] += x0.x * wc;
    sf[1] += x0.y * wa; cf[1] += x0.y * wc;
    sf[2] += x0.z * wa; cf[2] += x0.z * wc;
    sf[3] += x0.w * wa; cf[3] += x0.w * wc;
    sf[4] += x1.x * wa; cf[4] += x1.x * wc;
    sf[5] += x1.y * wa; cf[5] += x1.y * wc;
    sf[6] += x1.z * wa; cf[6] += x1.z * wc;
    sf[7] += x1.w * wa; cf[7] += x1.w * wc;
  }

  float* eself = emb + (size_t)b * (2 * C_) + c0;
  float* ecap  = eself + C_;
  *(float4*)(eself)     = make_float4(sf[0], sf[1], sf[2], sf[3]);
  *(float4*)(eself + 4) = make_float4(sf[4], sf[5], sf[6], sf[7]);
  *(float4*)(ecap)      = make_float4(cf[0], cf[1], cf[2], cf[3]);
  *(float4*)(ecap + 4)  = make_float4(cf[4], cf[5], cf[6], cf[7]);
}

// ---------------------------------------------------------------------------
extern "C" void kernel_launch(void* const* d_in, const int* in_sizes, int n_in,
                              void* d_out, int out_size, void* d_ws, size_t ws_size,
                              hipStream_t stream) {
  const float* img    = (const float*)d_in[0];   // (B, AREA, C)
  const float* att    = (const float*)d_in[1];   // (B, AREA, ATT)
  const float* hidden = (const float*)d_in[2];   // (1, B, RNN)
  const float* acap   = (const float*)d_in[3];   // (B, AREA)
  const float* W1     = (const float*)d_in[4];   // (ATT, RNN)
  const float* b1     = (const float*)d_in[5];   // (ATT,)
  const float* W2     = (const float*)d_in[6];   // (1, ATT)
  const float* b2     = (const float*)d_in[7];   // (1,)
  const float* W3     = (const float*)d_in[8];   // (C, 2C)
  const float* b3     = (const float*)d_in[9];   // (C,)

  float* out_emb = (float*)d_out;                // (B, C)
  float* out_a   = out_emb + (size_t)B_ * C_;    // (B, AREA)

  // Workspace layout
  float* ws     = (float*)d_ws;
  float* emb    = ws;                                   // B * 2C
  float* z1     = emb + (size_t)B_ * 2 * C_;            // B * ATT
  float* logits = z1 + (size_t)B_ * ATT;                // B * AREA

  // 1) z1 = hidden @ W1^T + b1   (M=256, N=512, K=512) -> 8x8 = 64 waves
  {
    const int waves = (B_ / 32) * (ATT / 64);
    wmma_gemm_atb<<<waves / 8, 256, 0, stream>>>(
        hidden, W1, b1, z1, ATT, RNN, ATT / 64, /*relu=*/0);
  }

  // 2) logits = tanh(att + z1) . W2 + b2   (one wave per (b,s) row)
  {
    const int rows = B_ * AREA;                          // 50176
    logits_kernel<<<rows / 8, 256, 0, stream>>>(att, z1, W2, b2, logits);
  }

  // 3) softmax over spatial dim -> a (written directly to output slot 2)
  softmax_kernel<<<B_, 256, 0, stream>>>(logits, out_a);

  // 4) dual weighted pooling -> emb (B, 2C)
  pool_kernel<<<B_, 256, 0, stream>>>(img, out_a, acap, emb);

  // 5) out = relu(emb @ W3^T + b3)   (M=256, N=2048, K=4096) -> 8x32 = 256 waves
  {
    const int waves = (B_ / 32) * (C_ / 64);
    wmma_gemm_atb<<<waves / 8, 256, 0, stream>>>(
        emb, W3, b3, out_emb, C_, 2 * C_, C_ / 64, /*relu=*/1);
  }
}